// TGCN2_33758442946925
// MI455X (gfx1250) — compile-verified
//
#include <hip/hip_runtime.h>
#include <math.h>

typedef __attribute__((ext_vector_type(2))) float v2f;
typedef __attribute__((ext_vector_type(8))) float v8f;

__device__ __forceinline__ float sigf(float v) { return 1.0f / (1.0f + expf(-v)); }

// ---------------- degree / norm ----------------
__global__ void k_deg_init(float* deg, int n) {
  int i = blockIdx.x * blockDim.x + threadIdx.x;
  if (i < n) deg[i] = 1.0f;  // self-loop weight
}
__global__ void k_deg_accum(const int* __restrict__ col, const float* __restrict__ w,
                            float* __restrict__ deg, int e) {
  int i = blockIdx.x * blockDim.x + threadIdx.x;
  if (i < e) atomicAdd(&deg[col[i]], w[i]);
}
__global__ void k_deg_fin(float* deg, int n) {
  int i = blockIdx.x * blockDim.x + threadIdx.x;
  if (i < n) {
    float d = deg[i];
    deg[i] = (d > 0.0f) ? rsqrtf(fmaxf(d, 1e-12f)) : 0.0f;  // becomes dis[]
  }
}
__global__ void k_norm(const int* __restrict__ row, const int* __restrict__ col,
                       const float* __restrict__ w, const float* __restrict__ dis,
                       float* __restrict__ nrm, int e) {
  int i = blockIdx.x * blockDim.x + threadIdx.x;
  if (i < e) nrm[i] = dis[row[i]] * w[i] * dis[col[i]];
}

// ---------------- fused 3-gate GEMM1: [BN,16] x (16x32)x3 -> [BN,96], fp32 WMMA ----------------
__global__ __launch_bounds__(256) void k_gemm1(const float* __restrict__ x,
                                               const float* __restrict__ Wz,
                                               const float* __restrict__ Wr,
                                               const float* __restrict__ Wh,
                                               float* __restrict__ out, int rows) {
  int wid = (blockIdx.x * blockDim.x + threadIdx.x) >> 5;
  int lane = threadIdx.x & 31;
  int m0 = wid * 16;
  if (m0 >= rows) return;          // wave-uniform: EXEC all-ones inside
  int half = lane >> 4;            // 0: K pair {0,1}; 1: K pair {2,3}
  int l = lane & 15;

  // A tile 16x16 (K=16) as 4 k-blocks of 16x4
  v2f a[4];
  const float* xr = x + (size_t)(m0 + l) * 16;
#pragma unroll
  for (int kb = 0; kb < 4; kb++) {
    int kbase = kb * 4 + half * 2;
    a[kb].x = xr[kbase];
    a[kb].y = xr[kbase + 1];
  }
  const float* Ws[3] = {Wz, Wr, Wh};
#pragma unroll
  for (int g = 0; g < 3; g++) {
    const float* W = Ws[g];
#pragma unroll
    for (int t = 0; t < 2; t++) {
      v8f acc = {0.f, 0.f, 0.f, 0.f, 0.f, 0.f, 0.f, 0.f};
#pragma unroll
      for (int kb = 0; kb < 4; kb++) {
        int kbase = kb * 4 + half * 2;
        v2f b;
        b.x = W[kbase * 32 + t * 16 + l];
        b.y = W[(kbase + 1) * 32 + t * 16 + l];
        acc = __builtin_amdgcn_wmma_f32_16x16x4_f32(false, a[kb], false, b,
                                                    (short)0, acc, false, false);
      }
      float* o = out + (size_t)m0 * 96 + g * 32 + t * 16 + l;
#pragma unroll
      for (int i = 0; i < 8; i++) o[(size_t)(half * 8 + i) * 96] = acc[i];
    }
  }
}

// ---------------- fused 3-gate GEMM2: relu([BN,96] slice) x (32x32)x3 -> [BN,96], fp32 WMMA ----------------
__global__ __launch_bounds__(256) void k_gemm2(const float* __restrict__ in96,
                                               const float* __restrict__ Wz,
                                               const float* __restrict__ Wr,
                                               const float* __restrict__ Wh,
                                               float* __restrict__ out, int rows) {
  int wid = (blockIdx.x * blockDim.x + threadIdx.x) >> 5;
  int lane = threadIdx.x & 31;
  int m0 = wid * 16;
  if (m0 >= rows) return;
  int half = lane >> 4;
  int l = lane & 15;
  const float* Ws[3] = {Wz, Wr, Wh};
#pragma unroll
  for (int g = 0; g < 3; g++) {
    // A tile 16x32 (K=32), relu folded into load
    v2f a[8];
    const float* ar = in96 + (size_t)(m0 + l) * 96 + g * 32;
#pragma unroll
    for (int kb = 0; kb < 8; kb++) {
      int kbase = kb * 4 + half * 2;
      a[kb].x = fmaxf(ar[kbase], 0.0f);
      a[kb].y = fmaxf(ar[kbase + 1], 0.0f);
    }
    const float* W = Ws[g];
#pragma unroll
    for (int t = 0; t < 2; t++) {
      v8f acc = {0.f, 0.f, 0.f, 0.f, 0.f, 0.f, 0.f, 0.f};
#pragma unroll
      for (int kb = 0; kb < 8; kb++) {
        int kbase = kb * 4 + half * 2;
        v2f b;
        b.x = W[kbase * 32 + t * 16 + l];
        b.y = W[(kbase + 1) * 32 + t * 16 + l];
        acc = __builtin_amdgcn_wmma_f32_16x16x4_f32(false, a[kb], false, b,
                                                    (short)0, acc, false, false);
      }
      float* o = out + (size_t)m0 * 96 + g * 32 + t * 16 + l;
#pragma unroll
      for (int i = 0; i < 8; i++) o[(size_t)(half * 8 + i) * 96] = acc[i];
    }
  }
}

// ---------------- out init: self-loop contribution + bias ----------------
__global__ void k_init_out(const float* __restrict__ hw, const float* __restrict__ dis,
                           const float* __restrict__ bz, const float* __restrict__ br,
                           const float* __restrict__ bh, float* __restrict__ out,
                           int N, int total) {
  int i = blockIdx.x * blockDim.x + threadIdx.x;
  if (i >= total) return;
  int f = i % 96;
  int bn = i / 96;
  int node = bn % N;
  int g = f >> 5, j = f & 31;
  float d = dis[node];
  const float* bias = (g == 0) ? bz : (g == 1) ? br : bh;
  out[i] = hw[i] * d * d + bias[j];
}

// ---------------- edge scatter: one wave per edge, 96 floats x 2 batches ----------------
__global__ __launch_bounds__(256) void k_scatter(const float* __restrict__ src,
                                                 float* __restrict__ dst,
                                                 const int* __restrict__ row,
                                                 const int* __restrict__ col,
                                                 const float* __restrict__ nrm,
                                                 int E, int N) {
  int wid = (blockIdx.x * blockDim.x + threadIdx.x) >> 5;
  int lane = threadIdx.x & 31;
  if (wid >= E) return;
  int r = row[wid], c = col[wid];
  float nv = nrm[wid];
#pragma unroll
  for (int b = 0; b < 2; b++) {
    const float* s = src + (size_t)(b * N + r) * 96;
    float* d = dst + (size_t)(b * N + c) * 96;
#pragma unroll
    for (int j = 0; j < 3; j++) atomicAdd(&d[j * 32 + lane], s[j * 32 + lane] * nv);
  }
}

// ---------------- gating: Z,R,H~ per node; sigmoid folded into load ----------------
__global__ __launch_bounds__(256) void k_final(const float* __restrict__ F,  // [BN,96] pre-sigmoid
                                               const float* __restrict__ H,  // [BN,32]
                                               const float* __restrict__ Lz, const float* __restrict__ blz,
                                               const float* __restrict__ Lr, const float* __restrict__ blr,
                                               const float* __restrict__ Lh, const float* __restrict__ blh,
                                               float* __restrict__ out, int total) {
  __shared__ float sLz[64 * 32];
  __shared__ float sLr[64 * 32];
  __shared__ float sLh[64 * 32];
  for (int i = threadIdx.x; i < 2048; i += blockDim.x) {
    sLz[i] = Lz[i];
    sLr[i] = Lr[i];
    sLh[i] = Lh[i];
  }
  __syncthreads();
  int lane = threadIdx.x & 31;
  int wid = (blockIdx.x * blockDim.x + threadIdx.x) >> 5;
  int nwaves = (gridDim.x * blockDim.x) >> 5;
  for (int node = wid; node < total; node += nwaves) {
    const float* f = F + (size_t)node * 96;
    float fz = sigf(f[lane]);
    float fr = sigf(f[32 + lane]);
    float fh = sigf(f[64 + lane]);
    float h = H[(size_t)node * 32 + lane];
    float zp = blz[lane], rp = blr[lane];
#pragma unroll 8
    for (int k = 0; k < 32; k++) {
      float fzk = __shfl(fz, k, 32);
      float frk = __shfl(fr, k, 32);
      float hk = __shfl(h, k, 32);
      zp += fzk * sLz[k * 32 + lane] + hk * sLz[(32 + k) * 32 + lane];
      rp += frk * sLr[k * 32 + lane] + hk * sLr[(32 + k) * 32 + lane];
    }
    float Z = sigf(zp);
    float R = sigf(rp);
    float hr = h * R;
    float hp = blh[lane];
#pragma unroll 8
    for (int k = 0; k < 32; k++) {
      float fhk = __shfl(fh, k, 32);
      float hrk = __shfl(hr, k, 32);
      hp += fhk * sLh[k * 32 + lane] + hrk * sLh[(32 + k) * 32 + lane];
    }
    float Ht = tanhf(hp);
    out[(size_t)node * 32 + lane] = Z * h + (1.0f - Z) * Ht;
  }
}

extern "C" void kernel_launch(void* const* d_in, const int* in_sizes, int n_in,
                              void* d_out, int out_size, void* d_ws, size_t ws_size,
                              hipStream_t stream) {
  const float* x   = (const float*)d_in[0];
  const int*   ei  = (const int*)d_in[1];
  const float* H   = (const float*)d_in[2];
  const float* ew  = (const float*)d_in[3];
  const float* Wz1 = (const float*)d_in[4];
  const float* bz1 = (const float*)d_in[5];
  const float* Wz2 = (const float*)d_in[6];
  const float* bz2 = (const float*)d_in[7];
  const float* Lz  = (const float*)d_in[8];
  const float* blz = (const float*)d_in[9];
  const float* Wr1 = (const float*)d_in[10];
  const float* br1 = (const float*)d_in[11];
  const float* Wr2 = (const float*)d_in[12];
  const float* br2 = (const float*)d_in[13];
  const float* Lr  = (const float*)d_in[14];
  const float* blr = (const float*)d_in[15];
  const float* Wh1 = (const float*)d_in[16];
  const float* bh1 = (const float*)d_in[17];
  const float* Wh2 = (const float*)d_in[18];
  const float* bh2 = (const float*)d_in[19];
  const float* Lh  = (const float*)d_in[20];
  const float* blh = (const float*)d_in[21];

  const int E  = in_sizes[3];
  const int BN = in_sizes[2] / 32;  // B*N (B==2)
  const int N  = BN / 2;
  const int* row = ei;
  const int* col = ei + E;

  // workspace carve (L2-resident double buffers)
  size_t off = 0;
  auto carve = [&](size_t bytes) -> char* {
    char* p = (char*)d_ws + off;
    off = (off + bytes + 255) & ~(size_t)255;
    return p;
  };
  float* dis  = (float*)carve((size_t)N * 4);
  float* nrm  = (float*)carve((size_t)E * 4);
  float* bufA = (float*)carve((size_t)BN * 96 * 4);
  float* bufB = (float*)carve((size_t)BN * 96 * 4);
  (void)ws_size; (void)n_in;

  dim3 blk(256);
  int gN = (N + 255) / 256;
  int gE = (E + 255) / 256;
  int mtiles = (BN + 15) / 16;
  int gGemm = (mtiles * 32 + 255) / 256;
  int gInit = (BN * 96 + 255) / 256;
  int gScat = (int)(((long long)E * 32 + 255) / 256);

  // gcn_norm
  k_deg_init<<<gN, blk, 0, stream>>>(dis, N);
  k_deg_accum<<<gE, blk, 0, stream>>>(col, ew, dis, E);
  k_deg_fin<<<gN, blk, 0, stream>>>(dis, N);
  k_norm<<<gE, blk, 0, stream>>>(row, col, ew, dis, nrm, E);

  // layer 1 (3 gates fused)
  k_gemm1<<<gGemm, blk, 0, stream>>>(x, Wz1, Wr1, Wh1, bufA, BN);
  k_init_out<<<gInit, blk, 0, stream>>>(bufA, dis, bz1, br1, bh1, bufB, N, BN * 96);
  k_scatter<<<gScat, blk, 0, stream>>>(bufA, bufB, row, col, nrm, E, N);

  // layer 2 (relu folded into GEMM2 A-load)
  k_gemm2<<<gGemm, blk, 0, stream>>>(bufB, Wz2, Wr2, Wh2, bufA, BN);
  k_init_out<<<gInit, blk, 0, stream>>>(bufA, dis, bz2, br2, bh2, bufB, N, BN * 96);
  k_scatter<<<gScat, blk, 0, stream>>>(bufA, bufB, row, col, nrm, E, N);

  // GRU gating
  k_final<<<416, blk, 0, stream>>>(bufB, H, Lz, blz, Lr, blr, Lh, blh, (float*)d_out, BN);
}